// SpecialStackLayer_27006754357583
// MI455X (gfx1250) — compile-verified
//
#include <hip/hip_runtime.h>

#ifndef __has_builtin
#define __has_builtin(x) 0
#endif

// Problem shape (fixed by the reference).
static constexpr int kBatch   = 32;
static constexpr int kSeqLen  = 4096;
static constexpr int kMaxSent = 128;
static constexpr int kDim     = 1024;   // 4 KB per row in f32

typedef float __attribute__((ext_vector_type(4))) float4v;
typedef int   __attribute__((ext_vector_type(4))) int4v;

// Pointer types matching the async builtin signature reported by clang:
// param 0 is 'int __vector(4) * __device__' (global), LDS side is addrspace(3).
typedef __attribute__((address_space(1))) int4v* gb128_t;
typedef __attribute__((address_space(3))) int4v* sb128_t;

// gfx1250 async global<->LDS copy path (ASYNCcnt-tracked), when the toolchain
// declares the builtins. Fallback: direct B128 load -> nontemporal B128 store.
#if __has_builtin(__builtin_amdgcn_global_load_async_to_lds_b128) && \
    __has_builtin(__builtin_amdgcn_global_store_async_from_lds_b128)
#define GATHER_USE_ASYNC 1
#else
#define GATHER_USE_ASYNC 0
#endif

__device__ __forceinline__ void wait_asynccnt0() {
#if __has_builtin(__builtin_amdgcn_s_wait_asynccnt)
  __builtin_amdgcn_s_wait_asynccnt(0);
#else
  asm volatile("s_wait_asynccnt 0" ::: "memory");
#endif
}

// One block per gathered row: 256 threads x 16 B = 4 KB = one DIM-row of f32.
__global__ __launch_bounds__(256) void gather_rows_kernel(
    const float* __restrict__ hidden,   // [32, 4096, 1024]
    const int*   __restrict__ pos,      // [32, 128]
    float*       __restrict__ out)      // [32, 128, 1024]
{
  const int r = blockIdx.x;                  // 0 .. kBatch*kMaxSent-1
  const int b = r >> 7;                      // r / kMaxSent
  const int j = r & (kMaxSent - 1);          // r % kMaxSent

  int p = pos[b * kMaxSent + j];
  // Clip like jnp.take_along_axis under jit.
  p = p < 0 ? 0 : (p >= kSeqLen ? kSeqLen - 1 : p);

  const float* src = hidden + ((size_t)b * kSeqLen + (size_t)p) * kDim;
  float*       dst = out    + (size_t)r * kDim;
  const int    t   = threadIdx.x;            // lane's 16-byte chunk id

#if GATHER_USE_ASYNC
  // CDNA5 async path: global -> LDS -> global, 16 B per lane, per-wave
  // ordering only (each lane reuses exactly the LDS bytes it loaded, so
  // s_wait_asynccnt 0 is sufficient; no workgroup barrier needed).
  __shared__ __align__(16) float lds_row[kDim];   // 4 KB staging

  gb128_t gsrc = (gb128_t)(src + (size_t)t * 4);  // 4 floats = 16 B per lane
  sb128_t srow = (sb128_t)(lds_row + t * 4);
  __builtin_amdgcn_global_load_async_to_lds_b128(gsrc, srow, /*offset=*/0, /*cpol=*/0);
  wait_asynccnt0();                               // LDS now holds this wave's bytes

  gb128_t gdst = (gb128_t)(dst + (size_t)t * 4);
  __builtin_amdgcn_global_store_async_from_lds_b128(gdst, srow, /*offset=*/0, /*cpol=*/0);
  wait_asynccnt0();                               // drain before wave retires
#else
  // Direct path: coalesced global_load_b128, streaming (non-temporal) store —
  // output is written once and never re-read, keep it out of the caches.
  float4v v = ((const float4v*)src)[t];
  __builtin_nontemporal_store(v, (float4v*)dst + t);
#endif
}

extern "C" void kernel_launch(void* const* d_in, const int* in_sizes, int n_in,
                              void* d_out, int out_size, void* d_ws, size_t ws_size,
                              hipStream_t stream) {
  (void)in_sizes; (void)n_in; (void)out_size; (void)d_ws; (void)ws_size;

  const float* hidden = (const float*)d_in[0];  // f32 [32,4096,1024]
  const int*   pos    = (const int*)d_in[1];    // int  [32,128]
  float*       out    = (float*)d_out;          // f32 [32,128,1024]

  dim3 grid(kBatch * kMaxSent);   // 4096 rows, one block each
  dim3 block(256);                // 8 wave32s; 16 B per lane
  gather_rows_kernel<<<grid, block, 0, stream>>>(hidden, pos, out);
}